// HetGAT_18622978195880
// MI455X (gfx1250) — compile-verified
//
#include <hip/hip_runtime.h>
#include <math.h>

// HetGAT fused kernel for gfx1250 (MI455X), wave32.
//
// Reformulation:
//   vs[s,t] = W[s] @ a[t, 0:64]      (9 vectors of 64)
//   va[t]   = W[t] @ a[t,64:128]     (3 vectors of 64)
//   e[s,t,b,n] = h[b,0].vs[s,t] + h[b,n].va[t]
//   softmax per branch -> w[t,b,n] = sum_s attn[s,t,b,n]
//   w[t,b,0] = mask_self[t,b]                  (self keep-mask)
//   x[b, t*64+i] = sum_n w[t,b,n] * h[b,n,i]
//   out[b] = elu( x[b,:] @ Wcat )   where Wcat = W viewed as (192,64) row-major
// Final GEMM done with V_WMMA_F32_16X16X4_F32.

#define BATCH    4096
#define NNODES   22      // 1 self + 10 ally + 11 opp
#define NALLY    10
#define NOPP     11
#define D        64
#define NEG_INF  (-1000000.0f)
#define MB       16      // batches per block
#define NTHREADS 256

typedef __attribute__((ext_vector_type(2))) float v2f;
typedef __attribute__((ext_vector_type(8))) float v8f;

// ---------------- prep: vs / va into workspace -------------------------------
__global__ void hetgat_prep(const float* __restrict__ W,
                            const float* __restrict__ a,
                            float* __restrict__ ws) {
    const int d = threadIdx.x;            // 0..63, one output dim per thread
    if (d >= D) return;
    // vs[s*3+t][d] = sum_i W[s][i][d] * a[t][i]
    for (int s = 0; s < 3; ++s) {
        for (int t = 0; t < 3; ++t) {
            float acc = 0.f;
            for (int i = 0; i < D; ++i)
                acc += W[s * D * D + i * D + d] * a[t * 2 * D + i];
            ws[(s * 3 + t) * D + d] = acc;
        }
    }
    // va[t][d] = sum_i W[t][i][d] * a[t][64+i]
    for (int t = 0; t < 3; ++t) {
        float acc = 0.f;
        for (int i = 0; i < D; ++i)
            acc += W[t * D * D + i * D + d] * a[t * 2 * D + D + i];
        ws[9 * D + t * D + d] = acc;
    }
}

// ---------------- main fused kernel ------------------------------------------
__global__ __launch_bounds__(NTHREADS)
void hetgat_main(const float* __restrict__ h,
                 const unsigned char* __restrict__ mask,   // bool (3,B,22)
                 const float* __restrict__ W,
                 const float* __restrict__ ws,
                 float* __restrict__ out) {
    __shared__ float s_vs[9 * D];
    __shared__ float s_va[3 * D];
    __shared__ float s_mask[MB][3][NNODES];   // 1.0 where mask==True
    __shared__ float s_en[MB][3][NNODES];     // h[b,n].va[t]
    __shared__ float s_es[MB][9];             // h[b,0].vs[s*3+t]
    __shared__ float s_w[MB][3][NNODES];      // combined weights
    __shared__ float s_x[MB][3 * D];          // GEMM A tile (16 x 192)

    const int tid = threadIdx.x;
    const int blockBase = blockIdx.x * MB;

    // ---- Phase A: stage vs/va, masks; zero weights ----
    for (int i = tid; i < 12 * D; i += NTHREADS) {
        float v = ws[i];
        if (i < 9 * D) s_vs[i] = v; else s_va[i - 9 * D] = v;
    }
    for (int i = tid; i < MB * 3 * NNODES; i += NTHREADS) {
        int b = i / (3 * NNODES);
        int rem = i % (3 * NNODES);
        int t = rem / NNODES;
        int n = rem % NNODES;
        unsigned char m = mask[((size_t)t * BATCH + (blockBase + b)) * NNODES + n];
        s_mask[b][t][n] = m ? 1.f : 0.f;
        s_w[b][t][n] = 0.f;
    }
    __syncthreads();

    // ---- Phase B: all logits. 8-lane groups; one (b,n) row per iteration ----
    {
        const int gid = tid >> 3;     // 0..31 groups
        const int l8  = tid & 7;      // lane within group
        for (int r = gid; r < MB * NNODES; r += 32) {
            const int b = r / NNODES;
            const int n = r % NNODES;
            const size_t base = ((size_t)(blockBase + b) * NNODES + n) * D + l8 * 8;
            const float4 h0 = *(const float4*)(h + base);
            const float4 h1 = *(const float4*)(h + base + 4);
            #pragma unroll
            for (int t = 0; t < 3; ++t) {
                const float* vp = &s_va[t * D + l8 * 8];
                float p = h0.x * vp[0] + h0.y * vp[1] + h0.z * vp[2] + h0.w * vp[3]
                        + h1.x * vp[4] + h1.y * vp[5] + h1.z * vp[6] + h1.w * vp[7];
                p += __shfl_xor(p, 1);
                p += __shfl_xor(p, 2);
                p += __shfl_xor(p, 4);
                if (l8 == 0) s_en[b][t][n] = p;
            }
            if (n == 0) {   // group-uniform branch; all 8 lanes active together
                #pragma unroll
                for (int st = 0; st < 9; ++st) {
                    const float* vp = &s_vs[st * D + l8 * 8];
                    float p = h0.x * vp[0] + h0.y * vp[1] + h0.z * vp[2] + h0.w * vp[3]
                            + h1.x * vp[4] + h1.y * vp[5] + h1.z * vp[6] + h1.w * vp[7];
                    p += __shfl_xor(p, 1);
                    p += __shfl_xor(p, 2);
                    p += __shfl_xor(p, 4);
                    if (l8 == 0) s_es[b][st] = p;
                }
            }
        }
    }
    __syncthreads();

    // ---- Phase C: masked softmax per branch, combine over s into s_w ----
    {
        const int lane = tid & 31;
        const int hl = lane & 15;                 // lane within half-wave
        const int b = (tid >> 5) * 2 + (lane >> 4);
        #pragma unroll
        for (int br = 0; br < 2; ++br) {
            const int nCnt  = br ? NOPP : NALLY;
            const int nBase = br ? 1 + NALLY : 1;
            const int cnt = 9 * nCnt;             // 90 or 99 entries
            const int iters = (cnt + 15) / 16;    // 6 or 7 (constant after unroll)
            float pv[7]; int tv[7], nv[7];
            float mx = NEG_INF;
            for (int i = 0; i < iters; ++i) {
                const int k = hl + i * 16;
                float e = NEG_INF; int tt = 0, nn = nBase;
                if (k < cnt) {
                    const int s = k / (3 * nCnt);
                    tt = (k / nCnt) % 3;
                    nn = nBase + (k % nCnt);
                    if (s_mask[b][tt][nn] < 0.5f)
                        e = s_es[b][s * 3 + tt] + s_en[b][tt][nn];
                }
                pv[i] = e; tv[i] = tt; nv[i] = nn;
                mx = fmaxf(mx, e);
            }
            mx = fmaxf(mx, __shfl_xor(mx, 1));
            mx = fmaxf(mx, __shfl_xor(mx, 2));
            mx = fmaxf(mx, __shfl_xor(mx, 4));
            mx = fmaxf(mx, __shfl_xor(mx, 8));
            float sum = 0.f;
            for (int i = 0; i < iters; ++i) {
                float p = (pv[i] > 0.5f * NEG_INF) ? expf(pv[i] - mx) : 0.f;
                pv[i] = p;
                sum += p;
            }
            sum += __shfl_xor(sum, 1);
            sum += __shfl_xor(sum, 2);
            sum += __shfl_xor(sum, 4);
            sum += __shfl_xor(sum, 8);
            const float scale = (sum > 0.f) ? 1.f / sum : 0.f;
            for (int i = 0; i < iters; ++i) {
                const float v = pv[i] * scale;
                if (v > 0.f) atomicAdd(&s_w[b][tv[i]][nv[i]], v);   // ds_add_f32
            }
        }
        if (hl == 0) {   // self term: keep-mask as the n=0 weight
            #pragma unroll
            for (int t = 0; t < 3; ++t) s_w[b][t][0] = s_mask[b][t][0];
        }
    }
    __syncthreads();

    // ---- Phase D: x[b][t*64+d] = sum_n w[t][n] * h[b][n][d] ----
    #pragma unroll
    for (int it = 0; it < (MB * D) / NTHREADS; ++it) {   // 4 iterations
        const int p = tid + it * NTHREADS;
        const int b = p >> 6;
        const int d = p & 63;
        const size_t base = ((size_t)(blockBase + b) * NNODES) * D + d;
        float a0 = 0.f, a1 = 0.f, a2 = 0.f;
        #pragma unroll
        for (int n = 0; n < NNODES; ++n) {
            const float hv = h[base + (size_t)n * D];
            a0 += s_w[b][0][n] * hv;
            a1 += s_w[b][1][n] * hv;
            a2 += s_w[b][2][n] * hv;
        }
        s_x[b][d]         = a0;
        s_x[b][D + d]     = a1;
        s_x[b][2 * D + d] = a2;
    }
    __syncthreads();

    // ---- Phase E: (16 x 192) @ (192 x 64) via V_WMMA_F32_16X16X4_F32 ----
    {
        const int wave = tid >> 5;
        const int lane = tid & 31;
        if (wave < 4) {                      // wave-uniform: EXEC all-1s inside
            const int nt   = wave;           // N tile (16 cols each)
            const int row  = lane & 15;      // A: M index, B/D: N index
            const int koff = (lane < 16) ? 0 : 2;
            const int col  = nt * 16 + row;
            v8f c = {};
            for (int k0 = 0; k0 < 3 * D; k0 += 4) {   // 48 WMMA ops, K=192
                v2f av, bv;
                av[0] = s_x[row][k0 + koff];
                av[1] = s_x[row][k0 + koff + 1];
                bv[0] = W[(size_t)(k0 + koff)     * D + col];
                bv[1] = W[(size_t)(k0 + koff + 1) * D + col];
                c = __builtin_amdgcn_wmma_f32_16x16x4_f32(
                        false, av, false, bv, (short)0, c, false, false);
            }
            #pragma unroll
            for (int r = 0; r < 8; ++r) {    // C/D: VGPR r -> M=r (+8 hi half)
                const int m = r + ((lane < 16) ? 0 : 8);
                float v = c[r];
                v = (v > 0.f) ? v : expm1f(v);         // elu
                out[(size_t)(blockBase + m) * D + col] = v;
            }
        }
    }
}

// ---------------- launch -----------------------------------------------------
extern "C" void kernel_launch(void* const* d_in, const int* in_sizes, int n_in,
                              void* d_out, int out_size, void* d_ws, size_t ws_size,
                              hipStream_t stream) {
    (void)in_sizes; (void)n_in; (void)out_size; (void)ws_size;
    const float*         h    = (const float*)d_in[0];
    // d_in[1]=num_ally, d_in[2]=num_opp are compile-time constants here
    const unsigned char* mask = (const unsigned char*)d_in[3];   // jax bool = 1 byte
    const float*         W    = (const float*)d_in[4];
    const float*         a    = (const float*)d_in[5];
    float* out = (float*)d_out;
    float* ws  = (float*)d_ws;   // needs 12*64*4 = 3 KB

    hetgat_prep<<<1, 64, 0, stream>>>(W, a, ws);
    hetgat_main<<<BATCH / MB, NTHREADS, 0, stream>>>(h, mask, W, ws, out);
}